// GINLayer_53326313947256
// MI455X (gfx1250) — compile-verified
//
#include <hip/hip_runtime.h>
#include <hip/hip_bf16.h>

// GIN layer for MI455X (gfx1250), fp32 end-to-end using V_WMMA_F32_16X16X4_F32.
// N=50000 (= 3125*16 = 5^5 * 16), D=O=128, E=800000.

typedef __attribute__((ext_vector_type(2))) float v2f;
typedef __attribute__((ext_vector_type(8))) float v8f;

#define NN 50000
#define DD 128
#define EE 800000
#define ROWTILES (NN / 16)        // 3125
#define WAVES_PER_BLOCK 5         // 3125 / 5 = 625 blocks, no tail

// ---------------- zero the BN stat accumulators (512 floats) ----------------
__global__ void gin_zero_stats(float* __restrict__ stats) {
    stats[threadIdx.x] = 0.0f;   // launched with 512 threads
}

// ---------------- h0 = (1+eps) * x  (vectorized float4) ---------------------
__global__ void gin_init_scaled(const float* __restrict__ x,
                                const float* __restrict__ epsp,
                                float* __restrict__ h) {
    const float f = 1.0f + epsp[0];
    int i = blockIdx.x * blockDim.x + threadIdx.x;     // over N*D/4 float4s
    float4 v = ((const float4*)x)[i];
    v.x *= f; v.y *= f; v.z *= f; v.w *= f;
    ((float4*)h)[i] = v;
}

// ---------------- scatter-add: h[dst] += x[src], one wave per edge ----------
__global__ void gin_edge_scatter(const float* __restrict__ x,
                                 const int* __restrict__ src,
                                 const int* __restrict__ dst,
                                 float* __restrict__ h) {
    const int e    = blockIdx.x * 8 + (threadIdx.x >> 5);   // 8 edges / 256-thr block
    const int lane = threadIdx.x & 31;
    const int s = src[e];
    const int d = dst[e];
    float4 v = ((const float4*)(x + (size_t)s * DD))[lane]; // 32 lanes * 4 = 128
    float* p = h + (size_t)d * DD + lane * 4;
    atomicAdd(p + 0, v.x);
    atomicAdd(p + 1, v.y);
    atomicAdd(p + 2, v.z);
    atomicAdd(p + 3, v.w);
}

// ------- Y = H @ W^T + b, plus per-column sum / sumsq for BatchNorm ---------
// One wave computes a 16x128 strip of Y (one row tile x all 8 column tiles):
// per k-step 1 A-load feeds 8 WMMAs (A register reuse), H read exactly once.
// A layout (16x4 f32): lane half h, vgpr j -> K = 2h+j  => lane loads float2
//   A[row0 + (lane&15)][k + 2h .. k + 2h + 1]
// B layout (4x16) taken symmetric: lane loads float2 W[ct*16+(lane&15)][k+2h..]
// C/D layout: vgpr r, half h, n=lane&15 -> element (row0 + r + 8h, ct*16 + n)
__global__ void __launch_bounds__(32 * WAVES_PER_BLOCK)
gin_gemm_bias_stats(const float* __restrict__ H, const float* __restrict__ W,
                    const float* __restrict__ bias, float* __restrict__ Y,
                    float* __restrict__ ssum, float* __restrict__ ssq) {
    __shared__ float lsum[WAVES_PER_BLOCK][DD];
    __shared__ float lsq [WAVES_PER_BLOCK][DD];

    const int wave = threadIdx.x >> 5;
    const int lane = threadIdx.x & 31;
    const int rt   = blockIdx.x * WAVES_PER_BLOCK + wave;   // row tile, 0..3124
    const int row0 = rt * 16;
    const int mn   = lane & 15;                 // M index for A, N index for B/C
    const int half = lane >> 4;

    v8f acc[8];
#pragma unroll
    for (int ct = 0; ct < 8; ++ct) {
        const float bv = bias[ct * 16 + mn];    // bias depends only on column
        acc[ct] = (v8f){ bv, bv, bv, bv, bv, bv, bv, bv };
    }

    const float* aptr = H + (size_t)(row0 + mn) * DD + 2 * half;
    const float* bptr = W + (size_t)mn * DD + 2 * half;   // + ct*16*DD per col tile

#pragma unroll 2
    for (int k = 0; k < DD; k += 4) {
        v2f a = *(const v2f*)(aptr + k);
#pragma unroll
        for (int ct = 0; ct < 8; ++ct) {
            v2f b = *(const v2f*)(bptr + ct * 16 * DD + k);
            acc[ct] = __builtin_amdgcn_wmma_f32_16x16x4_f32(
                /*neg_a=*/false, a, /*neg_b=*/false, b,
                /*c_mod=*/(short)0, acc[ct], /*reuse_a=*/false, /*reuse_b=*/false);
        }
    }

    // store strip + per-lane partial column stats -> LDS
#pragma unroll
    for (int ct = 0; ct < 8; ++ct) {
        float s = 0.0f, s2 = 0.0f;
#pragma unroll
        for (int r = 0; r < 8; ++r) {
            float v = acc[ct][r];
            Y[(size_t)(row0 + r + 8 * half) * DD + ct * 16 + mn] = v;
            s  += v;
            s2 += v * v;
        }
        // fold the two half-wave row groups (same column) together
        s  += __shfl_down(s, 16);
        s2 += __shfl_down(s2, 16);
        if (half == 0) {
            lsum[wave][ct * 16 + mn] = s;
            lsq [wave][ct * 16 + mn] = s2;
        }
    }
    __syncthreads();

    // block-level reduce over the 5 waves, then 2 atomics per column
    const int tid = threadIdx.x;
    if (tid < DD) {
        float t1 = 0.0f, t2 = 0.0f;
#pragma unroll
        for (int w = 0; w < WAVES_PER_BLOCK; ++w) {
            t1 += lsum[w][tid];
            t2 += lsq [w][tid];
        }
        atomicAdd(&ssum[tid], t1);
        atomicAdd(&ssq [tid], t2);
    }
}

// ---------------- out = relu( (Y - m) * rsqrt(var+eps) * gamma + beta ) -----
__global__ void gin_bn_relu(const float* __restrict__ Y,
                            const float* __restrict__ ssum,
                            const float* __restrict__ ssq,
                            const float* __restrict__ gamma,
                            const float* __restrict__ beta,
                            float* __restrict__ out) {
    const int i  = blockIdx.x * blockDim.x + threadIdx.x;   // over N*D/4 float4s
    const int c0 = (i & 31) << 2;                           // column of component 0
    const float invN = 1.0f / (float)NN;
    float4 y = ((const float4*)Y)[i];
    float r[4] = { y.x, y.y, y.z, y.w };
#pragma unroll
    for (int j = 0; j < 4; ++j) {
        const int c = c0 + j;
        const float m    = ssum[c] * invN;
        const float var  = fmaf(-m, m, ssq[c] * invN);      // biased var
        const float rstd = rsqrtf(var + 1e-5f);
        r[j] = fmaxf((r[j] - m) * rstd * gamma[c] + beta[c], 0.0f);
    }
    ((float4*)out)[i] = make_float4(r[0], r[1], r[2], r[3]);
}

extern "C" void kernel_launch(void* const* d_in, const int* in_sizes, int n_in,
                              void* d_out, int out_size, void* d_ws, size_t ws_size,
                              hipStream_t stream) {
    (void)in_sizes; (void)n_in; (void)out_size; (void)ws_size;

    const float* x    = (const float*)d_in[0];
    const int*   ei   = (const int*)  d_in[1];   // (2, E): [0:E)=src, [E:2E)=dst
    //           batch = d_in[2] (unused: single graph, no pooling in reference)
    const float* epsp = (const float*)d_in[3];
    const float* W1   = (const float*)d_in[4];
    const float* b1   = (const float*)d_in[5];
    const float* g1   = (const float*)d_in[6];
    const float* be1  = (const float*)d_in[7];
    const float* W2   = (const float*)d_in[8];
    const float* b2   = (const float*)d_in[9];
    const float* g2   = (const float*)d_in[10];
    const float* be2  = (const float*)d_in[11];
    float* out = (float*)d_out;

    float* A     = (float*)d_ws;                 // N*D fp32 (h buffer)
    float* B     = A + (size_t)NN * DD;          // N*D fp32 (gemm out)
    float* stats = B + (size_t)NN * DD;          // 512 floats: sum1,sq1,sum2,sq2

    // 1) zero BN stat accumulators
    gin_zero_stats<<<1, 512, 0, stream>>>(stats);

    // 2) A = (1+eps)*x
    gin_init_scaled<<<(NN * DD / 4) / 256, 256, 0, stream>>>(x, epsp, A);

    // 3) A[dst] += x[src] over all edges
    gin_edge_scatter<<<EE / 8, 256, 0, stream>>>(x, ei, ei + EE, A);

    // 4) B = A @ W1^T + b1, stats1
    gin_gemm_bias_stats<<<ROWTILES / WAVES_PER_BLOCK, 32 * WAVES_PER_BLOCK, 0, stream>>>(
        A, W1, b1, B, stats, stats + 128);

    // 5) A = relu(bn(B))
    gin_bn_relu<<<(NN * DD / 4) / 256, 256, 0, stream>>>(
        B, stats, stats + 128, g1, be1, A);

    // 6) B = A @ W2^T + b2, stats2
    gin_gemm_bias_stats<<<ROWTILES / WAVES_PER_BLOCK, 32 * WAVES_PER_BLOCK, 0, stream>>>(
        A, W2, b2, B, stats + 256, stats + 384);

    // 7) out = relu(bn(B))
    gin_bn_relu<<<(NN * DD / 4) / 256, 256, 0, stream>>>(
        B, stats + 256, stats + 384, g2, be2, out);
}